// SingleSlicePtychographyModel_5059471475187
// MI455X (gfx1250) — compile-verified
//
#include <hip/hip_runtime.h>
#include <math.h>

// ---------------------------------------------------------------------------
// Ptychography forward model on gfx1250 (CDNA5).
// All 256-point FFTs are done as 16x16 four-step DFTs using
// V_WMMA_F32_16X16X4_F32 (f32 tensor op => bit-compatible precision with the
// complex64 reference FFT path). 32 WMMAs per 256-point complex FFT.
// ---------------------------------------------------------------------------

typedef __attribute__((ext_vector_type(2))) float v2f;   // WMMA A/B frag (16x4 f32)
typedef __attribute__((ext_vector_type(8))) float v8f;   // WMMA C/D frag (16x16 f32)

#define TPB    128      // 4 wave32 waves per block
#define CHUNK  8        // batch positions processed per pass launch
#define NBAT   128
#define MMODE  4

static __device__ __forceinline__ float fftfreq256(int i) {
  return (float)(i < 128 ? i : i - 256) * (1.0f / 256.0f);
}

static __device__ __forceinline__ v8f wmma4(v2f a, v2f b, v8f c) {
  // D = A(16x4) * B(4x16) + C, f32 — chained over K for 16x16x16 products.
  return __builtin_amdgcn_wmma_f32_16x16x4_f32(false, a, false, b, (short)0, c,
                                               false, false);
}

// DFT-16 matrix (forward convention, e^{-2pi i mk/16}) and 256-pt twiddles.
static __device__ __forceinline__ void fft_init_tables(float* f16r, float* f16i,
                                                       float* twr, float* twi) {
  const float PI2 = 6.283185307179586f;
  for (int i = threadIdx.x; i < 256; i += blockDim.x) {
    int m = i >> 4, k = i & 15;
    float a = -(PI2 / 16.0f) * (float)(m * k);
    f16r[i] = cosf(a);
    f16i[i] = sinf(a);
    float p = -(PI2 / 256.0f) * (float)i;
    twr[i] = cosf(p);
    twi[i] = sinf(p);
  }
  __syncthreads();
}

#define FFT_SHARED                                                   \
  __shared__ float s_f16r[256], s_f16i[256], s_twr[256], s_twi[256]; \
  __shared__ float s_hr[4][256], s_hi[4][256];                       \
  fft_init_tables(s_f16r, s_f16i, s_twr, s_twi);                     \
  const int lane = threadIdx.x & 31;                                 \
  const int wv   = threadIdx.x >> 5;

// One 256-point complex FFT of a logical row, executed by one wave32.
//   ld(t)            -> float2 element t (0..255) of the logical row
//   st(v, k, re, im) -> output element k (v = C-fragment slot, fixed lane map)
//   sgn = +1 forward (e^{-2pi i}), -1 inverse; scale applied at output.
template <class LD, class ST>
static __device__ __forceinline__ void fft256_row(
    int lane, float* hr, float* hi, const float* f16r, const float* f16i,
    const float* twr, const float* twi, float sgn, float scale, LD ld, ST st) {
  const int half = lane >> 4;   // ISA: lanes 16..31 carry K+2 / M+8
  const int l16  = lane & 15;

  // ---- Stage 1: G = F16 * X   (X[n1][n2] = row[16*n1+n2]) ----
  v8f gr = {}; v8f gi = {};
#pragma unroll
  for (int c = 0; c < 4; ++c) {
    const int k0 = 4 * c + 2 * half;           // this lane's K pair
    v2f ar, ai, nai, bxr, bxi;
    ar[0]  = f16r[l16 * 16 + k0];      ar[1]  = f16r[l16 * 16 + k0 + 1];
    ai[0]  = sgn * f16i[l16 * 16 + k0];
    ai[1]  = sgn * f16i[l16 * 16 + k0 + 1];
    nai[0] = -ai[0];                   nai[1] = -ai[1];
    float2 x0 = ld(k0 * 16 + l16);             // X[k0  ][l16]
    float2 x1 = ld((k0 + 1) * 16 + l16);       // X[k0+1][l16]
    bxr[0] = x0.x; bxr[1] = x1.x;
    bxi[0] = x0.y; bxi[1] = x1.y;
    gr = wmma4(ar,  bxr, gr);   // Re += Fr*Xr
    gr = wmma4(nai, bxi, gr);   // Re -= Fi*Xi
    gi = wmma4(ar,  bxi, gi);   // Im += Fr*Xi
    gi = wmma4(ai,  bxr, gi);   // Im += Fi*Xr
  }

  // ---- Twiddle H[k1][n2] = G * w256^{k1*n2}; C-layout -> LDS (A-layout src)
#pragma unroll
  for (int v = 0; v < 8; ++v) {
    int k1 = v + 8 * half;
    int p  = (k1 * l16) & 255;
    float tr = twr[p], ti = sgn * twi[p];
    float a = gr[v], b = gi[v];
    hr[k1 * 16 + l16] = a * tr - b * ti;
    hi[k1 * 16 + l16] = a * ti + b * tr;
  }
  __builtin_amdgcn_wave_barrier();

  // ---- Stage 2: Y = H * F16  (F16 symmetric) ----
  v8f yr = {}; v8f yi = {};
#pragma unroll
  for (int c = 0; c < 4; ++c) {
    const int k0 = 4 * c + 2 * half;
    v2f ar, ai, nai, br, bi;
    ar[0]  = hr[l16 * 16 + k0];  ar[1]  = hr[l16 * 16 + k0 + 1];
    ai[0]  = hi[l16 * 16 + k0];  ai[1]  = hi[l16 * 16 + k0 + 1];
    nai[0] = -ai[0];             nai[1] = -ai[1];
    br[0] = f16r[k0 * 16 + l16];       br[1] = f16r[(k0 + 1) * 16 + l16];
    bi[0] = sgn * f16i[k0 * 16 + l16]; bi[1] = sgn * f16i[(k0 + 1) * 16 + l16];
    yr = wmma4(ar,  br, yr);
    yr = wmma4(nai, bi, yr);
    yi = wmma4(ar,  bi, yi);
    yi = wmma4(ai,  br, yi);
  }
  __builtin_amdgcn_wave_barrier();   // hr/hi reused by next row

  // ---- Output: X[k1 + 16*k2] = Y[k1][k2] ----
#pragma unroll
  for (int v = 0; v < 8; ++v) {
    int k1 = v + 8 * half;
    int k  = k1 + 16 * l16;
    st(v, k, yr[v] * scale, yi[v] * scale);
  }
}

// ---------------------------------------------------------------------------
// Kernels
// ---------------------------------------------------------------------------

// EW = exp(i*(obj_re + i*obj_im)) = exp(-obj_im) * (cos(obj_re) + i sin(obj_re))
__global__ void __launch_bounds__(TPB) k_ew(const float* __restrict__ ore,
                                            const float* __restrict__ oim,
                                            float2* __restrict__ EW) {
  int i = blockIdx.x * TPB + threadIdx.x;
  float e = expf(-oim[i]);
  float s, c;
  sincosf(ore[i], &s, &c);
  EW[i] = make_float2(e * c, e * s);
}

// Probe fft2, row pass: PA[m][y][kx] = FFT_x(probe[m][y][:])
__global__ void __launch_bounds__(TPB) k_probe_rows(const float* __restrict__ pre,
                                                    const float* __restrict__ pim,
                                                    float2* __restrict__ PA) {
  FFT_SHARED;
  int rowbase = blockIdx.x * 16;
  for (int i = 0; i < 4; ++i) {
    int r = rowbase + wv * 4 + i;        // 0..1023
    int m = r >> 8, y = r & 255;
    const float* sr = pre + m * 65536 + y * 256;
    const float* si = pim + m * 65536 + y * 256;
    float2* dst = PA + m * 65536 + y * 256;
    fft256_row(lane, s_hr[wv], s_hi[wv], s_f16r, s_f16i, s_twr, s_twi, 1.0f, 1.0f,
        [&](int t) { return make_float2(sr[t], si[t]); },
        [&](int v, int k, float re, float im) { dst[k] = make_float2(re, im); });
  }
}

// Probe fft2, column pass: PFt[m][kx][ky] = FFT_y(PA[m][:][kx])  (transposed)
__global__ void __launch_bounds__(TPB) k_probe_cols(const float2* __restrict__ PA,
                                                    float2* __restrict__ PFt) {
  FFT_SHARED;
  int rowbase = blockIdx.x * 16;
  for (int i = 0; i < 4; ++i) {
    int r = rowbase + wv * 4 + i;
    int m = r >> 8, x = r & 255;
    const float2* src = PA + m * 65536;
    float2* dst = PFt + m * 65536 + x * 256;
    fft256_row(lane, s_hr[wv], s_hi[wv], s_f16r, s_f16i, s_twr, s_twi, 1.0f, 1.0f,
        [&](int t) { return src[t * 256 + x]; },
        [&](int v, int k, float re, float im) { dst[k] = make_float2(re, im); });
  }
}

// Pass1: T1[y][x'] = IFFT_x( probe_f[m][y][x] * ramp_b[y][x] ), scale 1/256
__global__ void __launch_bounds__(TPB) k_pass1(const float2* __restrict__ PFt,
                                               const float* __restrict__ dr,
                                               int b0, float2* __restrict__ T1) {
  FFT_SHARED;
  const float PI2 = 6.283185307179586f;
  int bloc = blockIdx.x >> 6;
  int b = b0 + bloc;
  int rowbase = (blockIdx.x & 63) * 16;
  float dr0 = dr[2 * b + 0], dr1 = dr[2 * b + 1];
  float2* T1b = T1 + (size_t)bloc * (MMODE * 65536);
  for (int i = 0; i < 4; ++i) {
    int r = rowbase + wv * 4 + i;
    int m = r >> 8, y = r & 255;
    const float2* src = PFt + m * 65536;   // [kx][ky]
    float2* dst = T1b + m * 65536 + y * 256;
    float kyv = fftfreq256(y);
    fft256_row(lane, s_hr[wv], s_hi[wv], s_f16r, s_f16i, s_twr, s_twi,
        -1.0f, 1.0f / 256.0f,
        [&](int t) {
          float2 vv = src[t * 256 + y];              // probe_f[m][y][t]
          float ph = PI2 * (dr0 * kyv + dr1 * fftfreq256(t));
          float s, c;
          sincosf(ph, &s, &c);                       // ramp = c - i*s
          return make_float2(vv.x * c + vv.y * s, vv.y * c - vv.x * s);
        },
        [&](int v, int k, float re, float im) { dst[k] = make_float2(re, im); });
  }
}

// Pass2: T2[x][y] = IFFT_y(T1[:][x]), scale 1/256   (probe_s, transposed)
__global__ void __launch_bounds__(TPB) k_pass2(const float2* __restrict__ T1,
                                               float2* __restrict__ T2) {
  FFT_SHARED;
  int bloc = blockIdx.x >> 6;
  int rowbase = (blockIdx.x & 63) * 16;
  for (int i = 0; i < 4; ++i) {
    int r = rowbase + wv * 4 + i;
    int m = r >> 8, x = r & 255;
    const float2* src = T1 + (size_t)bloc * (MMODE * 65536) + m * 65536;
    float2* dst = T2 + (size_t)bloc * (MMODE * 65536) + m * 65536 + x * 256;
    fft256_row(lane, s_hr[wv], s_hi[wv], s_f16r, s_f16i, s_twr, s_twi,
        -1.0f, 1.0f / 256.0f,
        [&](int t) { return src[t * 256 + x]; },
        [&](int v, int k, float re, float im) { dst[k] = make_float2(re, im); });
  }
}

// Pass3: T3[x][ky] = FFT_y( patch[y][x] * probe_sT[x][y] )
__global__ void __launch_bounds__(TPB) k_pass3(const float2* __restrict__ T2,
                                               const float2* __restrict__ EW,
                                               const int* __restrict__ pos,
                                               int b0, float2* __restrict__ T3) {
  FFT_SHARED;
  int bloc = blockIdx.x >> 6;
  int b = b0 + bloc;
  int rowbase = (blockIdx.x & 63) * 16;
  int py = pos[2 * b + 0], px = pos[2 * b + 1];
  for (int i = 0; i < 4; ++i) {
    int r = rowbase + wv * 4 + i;
    int m = r >> 8, x = r & 255;
    const float2* src = T2 + (size_t)bloc * (MMODE * 65536) + m * 65536 + x * 256;
    float2* dst = T3 + (size_t)bloc * (MMODE * 65536) + m * 65536 + x * 256;
    fft256_row(lane, s_hr[wv], s_hi[wv], s_f16r, s_f16i, s_twr, s_twi, 1.0f, 1.0f,
        [&](int t) {
          float2 z = src[t];
          float2 w = EW[(size_t)(py + t) * 1024 + (px + x)];
          return make_float2(z.x * w.x - z.y * w.y, z.x * w.y + z.y * w.x);
        },
        [&](int v, int k, float re, float im) { dst[k] = make_float2(re, im); });
  }
}

// Pass4: out[b][ky][kx] = sum_m |FFT_x(T3[:][ky])|^2 / 65536   (ortho norm^2)
__global__ void __launch_bounds__(TPB) k_pass4(const float2* __restrict__ T3,
                                               int b0, float* __restrict__ out) {
  FFT_SHARED;
  int bloc = blockIdx.x >> 4;
  int b = b0 + bloc;
  int rowbase = (blockIdx.x & 15) * 16;
  for (int i = 0; i < 4; ++i) {
    int R = rowbase + wv * 4 + i;          // ky
    float acc[8] = {0, 0, 0, 0, 0, 0, 0, 0};
    for (int m = 0; m < MMODE; ++m) {
      const float2* src = T3 + (size_t)bloc * (MMODE * 65536) + m * 65536;
      fft256_row(lane, s_hr[wv], s_hi[wv], s_f16r, s_f16i, s_twr, s_twi, 1.0f, 1.0f,
          [&](int t) { return src[t * 256 + R]; },
          [&](int v, int k, float re, float im) { acc[v] += re * re + im * im; });
    }
    float* dst = out + (size_t)b * 65536 + R * 256;
    const float inv = 1.0f / 65536.0f;
#pragma unroll
    for (int v = 0; v < 8; ++v) {
      int k = (v + 8 * (lane >> 4)) + 16 * (lane & 15);
      dst[k] = acc[v] * inv;
    }
  }
}

// ---------------------------------------------------------------------------
extern "C" void kernel_launch(void* const* d_in, const int* in_sizes, int n_in,
                              void* d_out, int out_size, void* d_ws, size_t ws_size,
                              hipStream_t stream) {
  (void)in_sizes; (void)n_in; (void)out_size; (void)ws_size;
  const float* obj_re   = (const float*)d_in[0];
  const float* obj_im   = (const float*)d_in[1];
  const float* probe_re = (const float*)d_in[2];
  const float* probe_im = (const float*)d_in[3];
  const float* dr       = (const float*)d_in[4];
  const int*   positions = (const int*)d_in[5];
  float* out = (float*)d_out;

  // Workspace layout (float2): EW 1024^2 | PFt 4*256^2 | TA | TB (CHUNK*4*256^2 each)
  float2* EW  = (float2*)d_ws;
  float2* PFt = EW + 1024 * 1024;
  float2* TA  = PFt + MMODE * 65536;
  float2* TB  = TA + (size_t)CHUNK * MMODE * 65536;

  k_ew<<<1024 * 1024 / TPB, TPB, 0, stream>>>(obj_re, obj_im, EW);
  k_probe_rows<<<64, TPB, 0, stream>>>(probe_re, probe_im, TA);  // TA as scratch
  k_probe_cols<<<64, TPB, 0, stream>>>(TA, PFt);

  for (int b0 = 0; b0 < NBAT; b0 += CHUNK) {
    k_pass1<<<CHUNK * 64, TPB, 0, stream>>>(PFt, dr, b0, TA);
    k_pass2<<<CHUNK * 64, TPB, 0, stream>>>(TA, TB);
    k_pass3<<<CHUNK * 64, TPB, 0, stream>>>(TB, EW, positions, b0, TA);
    k_pass4<<<CHUNK * 16, TPB, 0, stream>>>(TA, b0, out);
  }
}